// RESCAL_24111946400626
// MI455X (gfx1250) — compile-verified
//
#include <hip/hip_runtime.h>
#include <hip/hip_bf16.h>

typedef __attribute__((ext_vector_type(2))) float v2f;
typedef __attribute__((ext_vector_type(4))) float v4f;
typedef __attribute__((ext_vector_type(8))) float v8f;

// Generic C[M,N] = A[M,K] * B[N,K]^T  (+ bias[n] if bias != nullptr)
// One wave (32 threads) computes one 16x16 tile of C using V_WMMA_F32_16X16X4_F32.
//
// f32 WMMA VGPR layouts (CDNA5 ISA 7.12.2):
//   A 16x4 : lanes 0-15  -> (M=lane,    K=0 in v0, K=1 in v1)
//            lanes 16-31 -> (M=lane-16, K=2 in v0, K=3 in v1)
//   B 4x16 : same pattern with N in place of M (B fed column-major, i.e. rows of Bsrc)
//   C/D 16x16: VGPR v holds (M = v + 8*(lane>=16), N = lane&15)
__global__ void __launch_bounds__(32)
rescal_gemm_nt_wmma(const float* __restrict__ A,    // [M, K]
                    const float* __restrict__ B,    // [N, K]
                    const float* __restrict__ bias, // [N] or nullptr
                    float* __restrict__ C,          // [M, N]
                    int K, int N) {
    const int m0   = blockIdx.x * 16;
    const int n0   = blockIdx.y * 16;
    const int lane = threadIdx.x;          // 0..31, all active (WMMA needs EXEC==~0)
    const int r    = lane & 15;
    const int kh   = (lane >> 4) << 1;     // 0 for lanes 0-15, 2 for lanes 16-31

    const float* arow = A + (size_t)(m0 + r) * K + kh;
    const float* brow = B + (size_t)(n0 + r) * K + kh;

    v8f acc = {};
    for (int k = 0; k < K; k += 4) {
        v2f a = *(const v2f*)(arow + k);
        v2f b = *(const v2f*)(brow + k);
        // 8 args: (neg_a, A, neg_b, B, c_mod, C, reuse_a, reuse_b)
        acc = __builtin_amdgcn_wmma_f32_16x16x4_f32(false, a, false, b,
                                                    (short)0, acc, false, false);
    }

    const float bn  = bias ? bias[n0 + r] : 0.0f;
    const int   mhi = (lane >> 4) * 8;
    float* crow = C + (size_t)(m0 + mhi) * N + (n0 + r);
#pragma unroll
    for (int v = 0; v < 8; ++v) {
        crow[(size_t)v * N] = acc[v] + bn;
    }
}

// scores[row=i*R+k][j] = uT[k][i] * uT[k][j]; one block per output row of NE floats.
// uT is tiny (R*NE*4 = 256 KB) -> L2 resident; output is 512 MB -> non-temporal stores.
__global__ void __launch_bounds__(256)
rescal_scores_kernel(const float* __restrict__ uT, // [R, NE]
                     float* __restrict__ out,      // [NE*R, NE]
                     int NE, int R) {
    const int row = blockIdx.x;       // row = i*R + k
    const int k   = row % R;
    const int i   = row / R;
    const float      s   = uT[(size_t)k * NE + i];
    const v4f* __restrict__ src = (const v4f*)(uT + (size_t)k * NE);
    v4f*       __restrict__ dst = (v4f*)(out + (size_t)row * NE);
    const int nv = NE >> 2;
    for (int t = threadIdx.x; t < nv; t += blockDim.x) {
        v4f v = src[t];
        v4f w = v * s;
        __builtin_nontemporal_store(w, dst + t);
    }
}

extern "C" void kernel_launch(void* const* d_in, const int* in_sizes, int n_in,
                              void* d_out, int out_size, void* d_ws, size_t ws_size,
                              hipStream_t stream) {
    const float* entities  = (const float*)d_in[0]; // [n_ent, dim]
    const float* relations = (const float*)d_in[1]; // [n_rel, dim]
    const float* W         = (const float*)d_in[2]; // [dim, dim]
    const float* b         = (const float*)d_in[3]; // [dim]

    const int dim   = in_sizes[3];              // 512
    const int n_ent = in_sizes[0] / dim;        // 2048
    const int n_rel = in_sizes[1] / dim;        // 32

    float* e  = (float*)d_ws;                   // [n_ent, dim]
    float* r  = e + (size_t)n_ent * dim;        // [n_rel, dim]
    float* uT = r + (size_t)n_rel * dim;        // [n_rel, n_ent]

    // e = entities @ W^T + b
    rescal_gemm_nt_wmma<<<dim3(n_ent / 16, dim / 16), 32, 0, stream>>>(
        entities, W, b, e, dim, dim);
    // r = relations @ W^T + b
    rescal_gemm_nt_wmma<<<dim3(n_rel / 16, dim / 16), 32, 0, stream>>>(
        relations, W, b, r, dim, dim);
    // uT[k][i] = dot(r_k, e_i)  ->  uT = r @ e^T   [n_rel, n_ent]
    rescal_gemm_nt_wmma<<<dim3(n_rel / 16, n_ent / 16), 32, 0, stream>>>(
        r, e, nullptr, uT, dim, n_ent);
    // scores[i,k,j] = uT[k,i] * uT[k,j]
    rescal_scores_kernel<<<n_ent * n_rel, 256, 0, stream>>>(
        uT, (float*)d_out, n_ent, n_rel);
}